// ASGCLayer_26834955666032
// MI455X (gfx1250) — compile-verified
//
#include <hip/hip_runtime.h>

#define F 96

typedef float v2f __attribute__((ext_vector_type(2)));
typedef float v8f __attribute__((ext_vector_type(8)));

// ---------------- zero workspace ----------------
__global__ void zero_ws_kernel(float* __restrict__ p, int n) {
    int i = blockIdx.x * blockDim.x + threadIdx.x;
    int stride = gridDim.x * blockDim.x;
    for (; i < n; i += stride) p[i] = 0.0f;
}

// ---------------- degree: deg[dst[e]] += 1 ----------------
__global__ void degree_kernel(const int* __restrict__ dst, float* __restrict__ deg, int ne) {
    int i = blockIdx.x * blockDim.x + threadIdx.x;
    int stride = gridDim.x * blockDim.x;
    for (; i < ne; i += stride) {
        if (i + stride < ne) __builtin_prefetch(&dst[i + stride], 0, 0);
        unsafeAtomicAdd(&deg[dst[i]], 1.0f);
    }
}

// ---------------- norm = (max(deg,1))^-0.5, in place ----------------
__global__ void norm_kernel(float* __restrict__ dn, int n) {
    int i = blockIdx.x * blockDim.x + threadIdx.x;
    if (i < n) {
        float d = dn[i];
        d = fmaxf(d, 1.0f);
        dn[i] = 1.0f / sqrtf(d);
    }
}

// ---------------- edge scatter: hacc[dst] += features[src] * norm[src] ----------------
// One wave32 per edge; lane L handles columns L, L+32, L+64 (coalesced 128B/req).
__global__ void scatter_kernel(const float* __restrict__ feat,
                               const int* __restrict__ src,
                               const int* __restrict__ dst,
                               const float* __restrict__ nrm,
                               float* __restrict__ hacc, int ne) {
    const int lane   = threadIdx.x & 31;
    const int wavesb = blockDim.x >> 5;
    int e = blockIdx.x * wavesb + (threadIdx.x >> 5);
    const int wstride = gridDim.x * wavesb;
    for (; e < ne; e += wstride) {
        if (e + wstride < ne) {
            __builtin_prefetch(&src[e + wstride], 0, 0);
            __builtin_prefetch(&dst[e + wstride], 0, 0);
        }
        const int s = src[e];
        const int d = dst[e];
        const float ns = nrm[s];
        const float* __restrict__ fr = feat + (size_t)s * F;
        float* __restrict__ hr = hacc + (size_t)d * F;
#pragma unroll
        for (int j = 0; j < 3; ++j) {
            const int c = lane + 32 * j;
            unsafeAtomicAdd(&hr[c], fr[c] * ns);
        }
    }
}

// ---------------- finalize with V_WMMA_F32_16X16X4_F32 ----------------
// One wave32 handles 16 nodes. D = A x B accumulated over 48 K-chunks of 4:
//   A[m,k] = hs[node_m, k]   (hs = [features | hacc*norm])
//   B[k,n] = a_weight[k]  broadcast over all 16 columns
// => every D column holds dot(hs_m, a_weight). Exact f32 math.
// ISA 16x4 f32 A layout: lanes 0-15 -> (K0,K1) in v[0],v[1]; lanes 16-31 -> (K2,K3).
__global__ void finalize_kernel(const float* __restrict__ feat,
                                const float* __restrict__ initf,
                                const float* __restrict__ aw,
                                const float* __restrict__ nrm,
                                const float* __restrict__ hacc,
                                float* __restrict__ out, int n) {
    const int lane = threadIdx.x & 31;
    const long wave = (long)blockIdx.x * (blockDim.x >> 5) + (threadIdx.x >> 5);
    const long base = wave * 16;
    if (base >= n) return;               // whole wave exits -> EXEC all-ones for WMMA

    const int half = lane >> 4;          // 0: K pair (0,1), 1: K pair (2,3) of the chunk
    const int mrow = lane & 15;
    long node = base + mrow;
    if (node >= n) node = n - 1;         // clamp loads; stores masked below

    const float* __restrict__ frow = feat + node * (long)F;
    const float* __restrict__ hrow = hacc + node * (long)F;
    const float  nn_a = nrm[node];

    v8f c = {};
#pragma unroll
    for (int chunk = 0; chunk < 24; ++chunk) {        // K = 0..95 : features part
        const int kb = chunk * 4 + half * 2;
        v2f a, b;
        a.x = frow[kb];     a.y = frow[kb + 1];
        b.x = aw[kb];       b.y = aw[kb + 1];
        c = __builtin_amdgcn_wmma_f32_16x16x4_f32(false, a, false, b, (short)0, c, false, false);
    }
#pragma unroll
    for (int chunk = 0; chunk < 24; ++chunk) {        // K = 96..191 : h*norm part
        const int kb = chunk * 4 + half * 2;
        v2f a, b;
        a.x = hrow[kb] * nn_a;      a.y = hrow[kb + 1] * nn_a;
        b.x = aw[F + kb];           b.y = aw[F + kb + 1];
        c = __builtin_amdgcn_wmma_f32_16x16x4_f32(false, a, false, b, (short)0, c, false, false);
    }

    // sigmoid over the 8 accumulator rows (rows r / r+8 depending on lane half)
    float alpha8[8];
#pragma unroll
    for (int r = 0; r < 8; ++r) alpha8[r] = 1.0f / (1.0f + __expf(-c[r]));

    // out[m,:] = alpha_m * (hacc[m,:]*norm[m]) + initial[m,:]
#pragma unroll
    for (int m = 0; m < 16; ++m) {
        const float alpha = __shfl(alpha8[m & 7], (m & 8) ? 16 : 0, 32);
        const long nm = base + m;
        if (nm >= n) continue;
        const float nn = nrm[nm];
        const float* __restrict__ hr = hacc + nm * (long)F;
        const float* __restrict__ ir = initf + nm * (long)F;
        float* __restrict__ orow = out + nm * (long)F;
#pragma unroll
        for (int j = 0; j < 3; ++j) {
            const int col = lane + 32 * j;
            orow[col] = alpha * (hr[col] * nn) + ir[col];
        }
    }
}

extern "C" void kernel_launch(void* const* d_in, const int* in_sizes, int n_in,
                              void* d_out, int out_size, void* d_ws, size_t ws_size,
                              hipStream_t stream) {
    const float* feat  = (const float*)d_in[0];   // (n,96)
    const float* initf = (const float*)d_in[1];   // (n,96)
    const float* aw    = (const float*)d_in[2];   // (1,192)
    const int*   src   = (const int*)d_in[3];     // (ne,)
    const int*   dst   = (const int*)d_in[4];     // (ne,)
    float* out = (float*)d_out;

    const int n  = in_sizes[0] / F;               // 50000
    const int ne = in_sizes[3];                   // 800000

    float* deg  = (float*)d_ws;                   // n floats (becomes norm in-place)
    float* hacc = deg + n;                        // n*96 floats

    const int nzero = n + n * F;
    zero_ws_kernel<<<4096, 256, 0, stream>>>(deg, nzero);
    degree_kernel<<<2048, 256, 0, stream>>>(dst, deg, ne);
    norm_kernel<<<(n + 255) / 256, 256, 0, stream>>>(deg, n);
    scatter_kernel<<<8192, 256, 0, stream>>>(feat, src, dst, deg, hacc, ne);

    const int waves  = (n + 15) / 16;
    const int blocks = (waves + 7) / 8;           // 8 waves (256 threads) per block
    finalize_kernel<<<blocks, 256, 0, stream>>>(feat, initf, aw, deg, hacc, out, n);
}